// GCN_16286515986672
// MI455X (gfx1250) — compile-verified
//
#include <hip/hip_runtime.h>
#include <hip/hip_bf16.h>
#include <cstdint>

// ---------------------------------------------------------------------------
// GCN (3x GCNConv + linear head) for MI455X / gfx1250.
//   N=100000 nodes, E=3200000 edges, F: 128 -> 64 -> 32 -> 16 -> 1
// GEMMs: V_WMMA_F32_16X16X4_F32 (fp32 matrix pipe), A-tile + full W staged in
// LDS via GLOBAL_LOAD_ASYNC_TO_LDS_B128 (ASYNCcnt path, inline asm).
// Edge aggregation: float4 gather + global_atomic_add_f32 scatter; feature
// tables (<=25.6MB) stay resident in the 192MB L2, so HBM only streams the
// edge list and x.
// ---------------------------------------------------------------------------

typedef __attribute__((ext_vector_type(2))) float v2f;
typedef __attribute__((ext_vector_type(8))) float v8f;

#define NODES 100000
#define NEDGE 3200000

// Async copy of one 16-byte chunk: global -> LDS, tracked on ASYNCcnt.
__device__ __forceinline__ void async_copy_b128(uint32_t lds_off, const void* gptr) {
    asm volatile("global_load_async_to_lds_b128 %0, %1, off"
                 :: "v"(lds_off), "v"((uint64_t)(uintptr_t)gptr)
                 : "memory");
}
__device__ __forceinline__ void async_wait0() {
    asm volatile("s_wait_asynccnt 0x0" ::: "memory");
}

// ---- degree / normalization ------------------------------------------------

__global__ void gcn_init_deg(float* __restrict__ deg, int n) {
    int i = blockIdx.x * blockDim.x + threadIdx.x;
    if (i < n) deg[i] = 1.0f;  // self loop
}

__global__ void gcn_accum_deg(const long long* __restrict__ dst,
                              float* __restrict__ deg, int e) {
    int i = blockIdx.x * blockDim.x + threadIdx.x;
    if (i < e) atomicAdd(&deg[(int)dst[i]], 1.0f);
}

__global__ void gcn_rsqrt(float* __restrict__ dinv, int n) {
    int i = blockIdx.x * blockDim.x + threadIdx.x;
    if (i < n) dinv[i] = rsqrtf(dinv[i]);  // deg >= 1 always
}

// ---- fp32 WMMA GEMM: Y[n x FOUT] = X[n x K] @ W[K x FOUT] -----------------
// One wave per 16x16 output tile; FOUT/16 waves per block. The 16xK A-tile
// (rows are contiguous in X) and the entire KxFOUT W are staged into LDS with
// async b128 copies, then fed to the matrix pipe.
// WMMA f32 layouts (ISA 7.12.2):
//   A(16x4): lane L holds A[L%16][2*(L/16)+{0,1}]
//   B(4x16): lane L holds B[2*(L/16)+{0,1}][L%16]
//   C/D:     lane L, reg r holds C[r + 8*(L/16)][L%16]

template <int K, int FOUT>
__global__ void gcn_gemm_wmma(const float* __restrict__ X,
                              const float* __restrict__ W,
                              float* __restrict__ Y) {
    __shared__ float Ash[16 * K];
    __shared__ float Wsh[K * FOUT];
    const int tid  = threadIdx.x;
    const int row0 = blockIdx.x * 16;

    // ---- async stage: A tile (16*K contiguous floats) ----
    {
        const float* gA = X + (size_t)row0 * K;
        const uint32_t aBase = (uint32_t)(uintptr_t)Ash;   // LDS byte offset
        for (int i = tid; i < (16 * K) / 4; i += blockDim.x)
            async_copy_b128(aBase + 16u * i, gA + 4 * i);
        // ---- async stage: full W (K*FOUT contiguous floats) ----
        const uint32_t wBase = (uint32_t)(uintptr_t)Wsh;
        for (int i = tid; i < (K * FOUT) / 4; i += blockDim.x)
            async_copy_b128(wBase + 16u * i, W + 4 * i);
    }
    async_wait0();
    __syncthreads();

    const int wave = tid >> 5;
    const int lane = tid & 31;
    const int m    = lane & 15;       // tile row (A) / col (B,C)
    const int half = lane >> 4;       // K-pair selector
    const int n    = wave * 16 + (lane & 15);

    v8f acc = {};
    #pragma unroll
    for (int k = 0; k < K; k += 4) {
        v2f a, b;
        const int ka = k + 2 * half;
        a.x = Ash[m * K + ka + 0];
        a.y = Ash[m * K + ka + 1];
        b.x = Wsh[(ka + 0) * FOUT + n];
        b.y = Wsh[(ka + 1) * FOUT + n];
        acc = __builtin_amdgcn_wmma_f32_16x16x4_f32(
            /*neg_a=*/false, a, /*neg_b=*/false, b,
            /*c_mod=*/(short)0, acc, /*reuse_a=*/false, /*reuse_b=*/false);
    }

    #pragma unroll
    for (int r = 0; r < 8; ++r)
        Y[(size_t)(row0 + r + 8 * half) * FOUT + n] = acc[r];
}

// ---- self-loop init: out[i,:] = dinv[i]^2 * xw[i,:] ------------------------

template <int F>
__global__ void gcn_selfloop_init(const float* __restrict__ xw,
                                  const float* __restrict__ dinv,
                                  float* __restrict__ out, int n) {
    long long t = (long long)blockIdx.x * blockDim.x + threadIdx.x;
    long long total = (long long)n * F;
    if (t >= total) return;
    int i = (int)(t / F);
    float di = dinv[i];
    out[t] = di * di * xw[t];
}

// ---- edge scatter: out[dst,:] += dinv[src]*dinv[dst] * xw[src,:] ----------
// One thread per (edge, 4-feature chunk): float4 gather + 4 f32 atomics.

template <int F>
__global__ void gcn_edge_scatter(const long long* __restrict__ src,
                                 const long long* __restrict__ dst,
                                 const float* __restrict__ dinv,
                                 const float* __restrict__ xw,
                                 float* __restrict__ out, int e) {
    constexpr int CH = F / 4;  // power of two -> shifts
    long long t = (long long)blockIdx.x * blockDim.x + threadIdx.x;
    long long total = (long long)e * CH;
    if (t >= total) return;
    long long ei = t / CH;
    int c = (int)(t - ei * CH);
    int s = (int)src[ei];
    int d = (int)dst[ei];
    float norm = dinv[s] * dinv[d];
    const float4 v = reinterpret_cast<const float4*>(xw + (size_t)s * F)[c];
    float* o = out + (size_t)d * F + c * 4;
    atomicAdd(o + 0, norm * v.x);
    atomicAdd(o + 1, norm * v.y);
    atomicAdd(o + 2, norm * v.z);
    atomicAdd(o + 3, norm * v.w);
}

// ---- bias + relu -----------------------------------------------------------

template <int F>
__global__ void gcn_bias_relu(float* __restrict__ h,
                              const float* __restrict__ b, int n) {
    long long t = (long long)blockIdx.x * blockDim.x + threadIdx.x;
    long long total = (long long)n * F;
    if (t >= total) return;
    int f = (int)(t % F);
    float v = h[t] + b[f];
    h[t] = v > 0.0f ? v : 0.0f;
}

// ---- head: out[i] = dot(h[i,:16], w) + b ----------------------------------

__global__ void gcn_head(const float* __restrict__ h,
                         const float* __restrict__ w,
                         const float* __restrict__ b,
                         float* __restrict__ out, int n) {
    int i = blockIdx.x * blockDim.x + threadIdx.x;
    if (i >= n) return;
    float s = b[0];
    #pragma unroll
    for (int f = 0; f < 16; ++f) s += h[(size_t)i * 16 + f] * w[f];
    out[i] = s;
}

// ---------------------------------------------------------------------------

static inline unsigned cdiv(long long a, long long b) {
    return (unsigned)((a + b - 1) / b);
}

extern "C" void kernel_launch(void* const* d_in, const int* in_sizes, int n_in,
                              void* d_out, int out_size, void* d_ws, size_t ws_size,
                              hipStream_t stream) {
    const float*     x     = (const float*)d_in[0];
    const long long* eidx  = (const long long*)d_in[1];
    const float*     w1    = (const float*)d_in[2];
    const float*     b1    = (const float*)d_in[3];
    const float*     w2    = (const float*)d_in[4];
    const float*     b2    = (const float*)d_in[5];
    const float*     w3    = (const float*)d_in[6];
    const float*     b3    = (const float*)d_in[7];
    const float*     w_out = (const float*)d_in[8];
    const float*     b_out = (const float*)d_in[9];
    float*           out   = (float*)d_out;

    const int n = NODES, e = NEDGE;
    const long long E64 = e;
    const long long half = (in_sizes[1] / 2);     // = E
    const long long* src = eidx;
    const long long* dst = eidx + half;

    // workspace layout: dinv [N], bufA [N*64], bufB [N*64]
    char* ws = (char*)d_ws;
    float* dinv = (float*)ws;                      ws += ((size_t)n * 4 + 255) & ~255ull;
    float* bufA = (float*)ws;                      ws += ((size_t)n * 64 * 4 + 255) & ~255ull;
    float* bufB = (float*)ws;

    const int B = 256;

    // normalization (once; same graph for all layers)
    gcn_init_deg <<<cdiv(n, B), B, 0, stream>>>(dinv, n);
    gcn_accum_deg<<<cdiv(e, B), B, 0, stream>>>(dst, dinv, e);
    gcn_rsqrt    <<<cdiv(n, B), B, 0, stream>>>(dinv, n);

    // ---- layer 1: 128 -> 64 ----
    gcn_gemm_wmma<128, 64><<<n / 16, 32 * (64 / 16), 0, stream>>>(x, w1, bufA);
    gcn_selfloop_init<64><<<cdiv((long long)n * 64, B), B, 0, stream>>>(bufA, dinv, bufB, n);
    gcn_edge_scatter <64><<<cdiv(E64 * 16, B), B, 0, stream>>>(src, dst, dinv, bufA, bufB, e);
    gcn_bias_relu    <64><<<cdiv((long long)n * 64, B), B, 0, stream>>>(bufB, b1, n);

    // ---- layer 2: 64 -> 32 ----
    gcn_gemm_wmma<64, 32><<<n / 16, 32 * (32 / 16), 0, stream>>>(bufB, w2, bufA);
    gcn_selfloop_init<32><<<cdiv((long long)n * 32, B), B, 0, stream>>>(bufA, dinv, bufB, n);
    gcn_edge_scatter <32><<<cdiv(E64 * 8, B), B, 0, stream>>>(src, dst, dinv, bufA, bufB, e);
    gcn_bias_relu    <32><<<cdiv((long long)n * 32, B), B, 0, stream>>>(bufB, b2, n);

    // ---- layer 3: 32 -> 16 ----
    gcn_gemm_wmma<32, 16><<<n / 16, 32 * (16 / 16), 0, stream>>>(bufB, w3, bufA);
    gcn_selfloop_init<16><<<cdiv((long long)n * 16, B), B, 0, stream>>>(bufA, dinv, bufB, n);
    gcn_edge_scatter <16><<<cdiv(E64 * 4, B), B, 0, stream>>>(src, dst, dinv, bufA, bufB, e);
    gcn_bias_relu    <16><<<cdiv((long long)n * 16, B), B, 0, stream>>>(bufB, b3, n);

    // ---- head: 16 -> 1 ----
    gcn_head<<<cdiv(n, B), B, 0, stream>>>(bufB, w_out, b_out, out, n);
}